// distance_9216999817567
// MI455X (gfx1250) — compile-verified
//
#include <hip/hip_runtime.h>

// ---------------------------------------------------------------------------
// Min-image distance kernel (pure bandwidth-bound gather/scatter).
//
// Inputs (setup_inputs order):
//   d_in[0] xyz  : float32 [N,3]   (1.2 MB -> stays resident in L2)
//   d_in[1] src  : int32   [E]
//   d_in[2] dst  : int32   [E]
//   d_in[3] cell : float32 [1,3]
// Output (concatenated): dis [E] floats, then dis_vec2*mask [E,3] floats.
// ---------------------------------------------------------------------------

typedef float  f4 __attribute__((ext_vector_type(4)));
typedef int    i4 __attribute__((ext_vector_type(4)));

#define EPS_SIGN 1e-8f
#define EPS_NORM 1e-9f

// Per-component min-image + mask, exactly mirroring the reference math.
__device__ __forceinline__ float edge_component(float d, float c) {
    float a     = fabsf(d);
    float m     = fminf(c - a, a);                    // min-image magnitude (may be <0)
    float mask2 = (fabsf(m) == a) ? 1.0f : -1.0f;     // exact equality, as in reference
    float de    = d + EPS_SIGN;
    float sgn   = de / fabsf(de);                     // epsilon-stabilized sign
    return m * mask2 * sgn;
}

__device__ __forceinline__ void edge_full(const float* __restrict__ xyz,
                                          int s, int d,
                                          float cx, float cy, float cz,
                                          float& dis, float& vx, float& vy, float& vz) {
    const float* ps = xyz + 3 * s;   // compiler merges into global_load_b96
    const float* pd = xyz + 3 * d;
    float dx = ps[0] - pd[0];
    float dy = ps[1] - pd[1];
    float dz = ps[2] - pd[2];

    vx = edge_component(dx, cx);
    vy = edge_component(dy, cy);
    vz = edge_component(dz, cz);

    // Norm of the ORIGINAL displacement, each component + 1e-9 (per source).
    float ex = dx + EPS_NORM, ey = dy + EPS_NORM, ez = dz + EPS_NORM;
    dis = sqrtf(ex * ex + ey * ey + ez * ez);
}

__global__ __launch_bounds__(256)
void minimage_edges_kernel(const float* __restrict__ xyz,
                           const int*   __restrict__ src,
                           const int*   __restrict__ dst,
                           const float* __restrict__ cell,
                           float* __restrict__ out_dis,
                           float* __restrict__ out_vec,
                           int n_edges) {
    const float cx = cell[0];
    const float cy = cell[1];
    const float cz = cell[2];

    const int t  = blockIdx.x * blockDim.x + threadIdx.x;
    const int e0 = t * 4;
    if (e0 >= n_edges) return;

    // gfx1250 global_prefetch_b8: pull index cachelines ~64KB ahead of the
    // latency-exposed gather loop. Speculative; OOB translation is dropped.
    {
        const int pf = e0 + 16384;
        if (pf < n_edges) {
            __builtin_prefetch(src + pf, 0, 1);
            __builtin_prefetch(dst + pf, 0, 1);
        }
    }

    if (((n_edges & 3) == 0) && (e0 + 4 <= n_edges)) {
        // ---- fast path: 4 edges, b128 non-temporal streaming I/O ----
        i4 s4 = __builtin_nontemporal_load((const i4*)(src + e0));
        i4 d4 = __builtin_nontemporal_load((const i4*)(dst + e0));

        float dis[4], v[12];
#pragma unroll
        for (int j = 0; j < 4; ++j) {
            edge_full(xyz, s4[j], d4[j], cx, cy, cz,
                      dis[j], v[3 * j + 0], v[3 * j + 1], v[3 * j + 2]);
        }

        f4 dq = {dis[0], dis[1], dis[2], dis[3]};
        __builtin_nontemporal_store(dq, (f4*)(out_dis + e0));

        // out_vec base is 16B-aligned because n_edges % 4 == 0 on this path.
        f4 v0 = {v[0], v[1], v[2],  v[3]};
        f4 v1 = {v[4], v[5], v[6],  v[7]};
        f4 v2 = {v[8], v[9], v[10], v[11]};
        f4* vp = (f4*)(out_vec + 3 * (size_t)e0);
        __builtin_nontemporal_store(v0, vp + 0);
        __builtin_nontemporal_store(v1, vp + 1);
        __builtin_nontemporal_store(v2, vp + 2);
    } else {
        // ---- tail path: scalar, alignment-safe ----
        for (int e = e0; e < n_edges && e < e0 + 4; ++e) {
            float dis, vx, vy, vz;
            edge_full(xyz, src[e], dst[e], cx, cy, cz, dis, vx, vy, vz);
            __builtin_nontemporal_store(dis, out_dis + e);
            __builtin_nontemporal_store(vx, out_vec + 3 * (size_t)e + 0);
            __builtin_nontemporal_store(vy, out_vec + 3 * (size_t)e + 1);
            __builtin_nontemporal_store(vz, out_vec + 3 * (size_t)e + 2);
        }
    }
}

extern "C" void kernel_launch(void* const* d_in, const int* in_sizes, int n_in,
                              void* d_out, int out_size, void* d_ws, size_t ws_size,
                              hipStream_t stream) {
    const float* xyz  = (const float*)d_in[0];
    const int*   src  = (const int*)  d_in[1];
    const int*   dst  = (const int*)  d_in[2];
    const float* cell = (const float*)d_in[3];

    const int n_edges = in_sizes[1];          // E
    float* out_dis = (float*)d_out;           // [E]
    float* out_vec = (float*)d_out + n_edges; // [E,3] flattened

    const int threads = 256;
    const int work    = (n_edges + 3) / 4;    // 4 edges per thread
    const int blocks  = (work + threads - 1) / threads;

    minimage_edges_kernel<<<blocks, threads, 0, stream>>>(
        xyz, src, dst, cell, out_dis, out_vec, n_edges);
}